// RegularizationLoss_72962904425129
// MI455X (gfx1250) — compile-verified
//
#include <hip/hip_runtime.h>

#define TPB      256
#define NBLOCKS  1024
#define WPB      (TPB / 32)            // waves per block (wave32)
#define NWAVES   (NBLOCKS * WPB)       // total waves in grid
#define TILE_B   32                    // batches per tile (one per lane)
#define TILE_FLOATS (TILE_B * 18)      // 576 floats
#define TILE_BYTES  (TILE_FLOATS * 4)  // 2304 bytes

// ||M M^T - I||_F^2 for a row-major 3x3 matrix r[9]
__device__ __forceinline__ float frob_gram(const float* r) {
    float d00 = r[0]*r[0] + r[1]*r[1] + r[2]*r[2];
    float d11 = r[3]*r[3] + r[4]*r[4] + r[5]*r[5];
    float d22 = r[6]*r[6] + r[7]*r[7] + r[8]*r[8];
    float d01 = r[0]*r[3] + r[1]*r[4] + r[2]*r[5];
    float d02 = r[0]*r[6] + r[1]*r[7] + r[2]*r[8];
    float d12 = r[3]*r[6] + r[4]*r[7] + r[5]*r[8];
    float a = d00 - 1.f, b = d11 - 1.f, c = d22 - 1.f;
    return a*a + b*b + c*c + 2.f*(d01*d01 + d02*d02 + d12*d12);
}

// Stage one 2304-byte tile (32 batches) Global -> LDS via the CDNA5 async
// copy path. 4 x b128 + 1 x b64 per lane covers 32*72B exactly, fully
// coalesced (the wave covers a contiguous 2304B span; every cacheline byte
// is consumed). Default RT temporal hint on purpose: the 144MB input fits
// in the 192MB L2, so graph-replay passes can run out of L2.
__device__ __forceinline__ void async_tile(const void* gbase, unsigned ldsbase, int lane) {
    const char* g = (const char*)gbase;
#pragma unroll
    for (int k = 0; k < 4; ++k) {
        asm volatile("global_load_async_to_lds_b128 %0, %1, off"
                     :: "v"(ldsbase + (unsigned)(k*512 + lane*16)),
                        "v"(g + k*512 + lane*16)
                     : "memory");
    }
    asm volatile("global_load_async_to_lds_b64 %0, %1, off"
                 :: "v"(ldsbase + (unsigned)(2048 + lane*8)),
                    "v"(g + 2048 + lane*8)
                 : "memory");
}

// Read this lane's 18 floats from LDS (8B-aligned; lane stride = 18 dwords
// => 18*(dl) % 64 == 0 only when dl % 32 == 0, so all 32 lanes hit distinct
// banks: conflict-free) and evaluate both 3x3 matrices.
__device__ __forceinline__ float tile_batch_term(const float* p) {
    float r[18];
#pragma unroll
    for (int j = 0; j < 9; ++j) {
        float2 v = *(const float2*)(p + 2*j);
        r[2*j] = v.x; r[2*j+1] = v.y;
    }
    return frob_gram(r) + frob_gram(r + 9);
}

__global__ __launch_bounds__(TPB)
void reg_loss_partials(const float* __restrict__ in, double* __restrict__ part,
                       int nBatches, int nTiles) {
    __shared__ float  stage[WPB * 2 * TILE_FLOATS];  // double-buffered per wave
    __shared__ double red[TPB];

    const int lane = threadIdx.x & 31;
    const int w    = threadIdx.x >> 5;
    const int wid  = blockIdx.x * WPB + w;

    float* wstage = &stage[w * 2 * TILE_FLOATS];
    // Low 32 bits of a generic pointer to LDS == workgroup-relative LDS byte address.
    const unsigned wbase0 = (unsigned)(size_t)(void*)wstage;
    const unsigned wbase1 = wbase0 + TILE_BYTES;

    double acc = 0.0;

    if (wid < nTiles) {
        // prime buffer 0
        async_tile(in + (size_t)wid * TILE_FLOATS, wbase0, lane);
        int buf = 0;
        int t   = wid;
        // steady state: branchless issue-next / wait-oldest-5 / compute / flip
        for (int tn = wid + NWAVES; tn < nTiles; tn += NWAVES) {
            async_tile(in + (size_t)tn * TILE_FLOATS, buf ? wbase0 : wbase1, lane);
            // async loads complete in order: <=5 outstanding means the 5 ops
            // of the current tile have landed in LDS
            asm volatile("s_wait_asynccnt 5" ::: "memory");
            acc += (double)tile_batch_term(wstage + buf * TILE_FLOATS + lane * 18);
            buf ^= 1;
            t = tn;
        }
        // peeled last tile
        asm volatile("s_wait_asynccnt 0" ::: "memory");
        acc += (double)tile_batch_term(wstage + buf * TILE_FLOATS + lane * 18);
        (void)t;
    }

    // tail (nBatches not a multiple of 32): plain global reads, wave 0 only
    if (wid == 0) {
        const int rem = nBatches - nTiles * TILE_B;
        if (lane < rem) {
            const float* p = in + (size_t)(nTiles * TILE_B + lane) * 18;
            float r[18];
#pragma unroll
            for (int j = 0; j < 18; ++j) r[j] = p[j];
            acc += (double)(frob_gram(r) + frob_gram(r + 9));
        }
    }

    // deterministic block tree reduction (fixed order => bitwise reproducible)
    red[threadIdx.x] = acc;
    __syncthreads();
#pragma unroll
    for (int s = TPB / 2; s > 0; s >>= 1) {
        if (threadIdx.x < s) red[threadIdx.x] += red[threadIdx.x + s];
        __syncthreads();
    }
    if (threadIdx.x == 0) part[blockIdx.x] = red[0];
}

__global__ __launch_bounds__(TPB)
void reg_loss_finalize(const double* __restrict__ part, int n,
                       float* __restrict__ out, int nBatches) {
    __shared__ double red[TPB];
    double a = 0.0;
    for (int i = threadIdx.x; i < n; i += TPB) a += part[i];  // fixed order
    red[threadIdx.x] = a;
    __syncthreads();
#pragma unroll
    for (int s = TPB / 2; s > 0; s >>= 1) {
        if (threadIdx.x < s) red[threadIdx.x] += red[threadIdx.x + s];
        __syncthreads();
    }
    if (threadIdx.x == 0) out[0] = (float)(red[0] / (double)nBatches);
}

extern "C" void kernel_launch(void* const* d_in, const int* in_sizes, int n_in,
                              void* d_out, int out_size, void* d_ws, size_t ws_size,
                              hipStream_t stream) {
    const float* in = (const float*)d_in[0];
    const int total    = in_sizes[0];
    const int nBatches = total / 18;
    const int nTiles   = nBatches / TILE_B;
    double* part = (double*)d_ws;   // NBLOCKS doubles = 8 KB scratch

    reg_loss_partials<<<NBLOCKS, TPB, 0, stream>>>(in, part, nBatches, nTiles);
    reg_loss_finalize<<<1, TPB, 0, stream>>>(part, NBLOCKS, (float*)d_out, nBatches);
}